// LeakGAT_15874199126615
// MI455X (gfx1250) — compile-verified
//
#include <hip/hip_runtime.h>
#include <hip/hip_bf16.h>
#include <math.h>

#define D1 128          // HEADS*HID
#define HEADS 4
#define HID 32

typedef __attribute__((ext_vector_type(2))) float v2f;
typedef __attribute__((ext_vector_type(8))) float v8f;

// ---------- helpers ----------
__device__ __forceinline__ unsigned fenc(float x) {
    unsigned u = __float_as_uint(x);
    return (u & 0x80000000u) ? ~u : (u | 0x80000000u);   // order-preserving map
}
__device__ __forceinline__ float fdec(unsigned u) {
    unsigned v = (u & 0x80000000u) ? (u & 0x7FFFFFFFu) : ~u;
    return __uint_as_float(v);
}
__device__ __forceinline__ void get_edge(const int* __restrict__ ei, int E, int e,
                                         int& s, int& d) {
    if (e < E) { s = ei[e]; d = ei[E + e]; }
    else       { s = e - E; d = e - E; }                 // implicit self-loop
}

// ---------- layer-1 GEMM: [n,3] x [3,128] ----------
__global__ __launch_bounds__(256) void gemm3_kernel(const float* __restrict__ X,
                                                    const float* __restrict__ W,
                                                    float* __restrict__ XP, int n) {
    int i = blockIdx.x * 256 + threadIdx.x;
    if (i >= n * D1) return;
    int node = i >> 7, j = i & 127;
    const float* xr = X + (size_t)node * 3;
    XP[i] = xr[0] * W[j] + xr[1] * W[D1 + j] + xr[2] * W[2 * D1 + j];
}

// ---------- layers 2/3 GEMM: [n,128] x [128,128] via V_WMMA_F32_16X16X4_F32 ----------
// W staged in LDS via GLOBAL_LOAD_ASYNC_TO_LDS_B128 (ASYNCcnt path, no VGPR round trip)
__global__ __launch_bounds__(256) void gemm128_wmma_kernel(const float* __restrict__ H,
                                                           const float* __restrict__ W,
                                                           float* __restrict__ XP, int n) {
    __shared__ float Wl[D1 * D1];                        // 64 KB of the 320 KB WGP LDS
    const int t = threadIdx.x;

    // async stage of W: 4096 x b128, 16 per thread, no VGPR data movement
    {
        unsigned ldsBase = (unsigned)(uintptr_t)(&Wl[0]);  // LDS aperture: low 32b = LDS addr
        unsigned long long gBase = (unsigned long long)(uintptr_t)W;
        #pragma unroll
        for (int i = 0; i < 16; ++i) {
            unsigned idx = (unsigned)t + (unsigned)i * 256u;     // float4 index
            unsigned lds_off = ldsBase + idx * 16u;
            unsigned long long gaddr = gBase + (unsigned long long)idx * 16u;
            asm volatile("global_load_async_to_lds_b128 %0, %1, off"
                         :: "v"(lds_off), "v"(gaddr)
                         : "memory");
        }
        asm volatile("s_wait_asynccnt 0" ::: "memory");
    }
    __syncthreads();

    const int wave    = t >> 5;
    const int lane    = t & 31;
    const int rowBase = blockIdx.x * 128 + wave * 16;    // 8 waves x 16 rows per block
    if (rowBase >= n) return;                            // wave-uniform; n % 16 == 0

    const int row     = rowBase + (lane & 15);           // A frag: M = lane&15
    const int kOff    = (lane >> 4) * 2;                 // lanes 16-31 carry K+2,K+3
    const int colLane = lane & 15;

    // Preload full-K A fragments: 32 steps of (K=4) -> 64 VGPRs
    v2f a[32];
    const float* hrow = H + (size_t)row * D1 + kOff;
    #pragma unroll
    for (int s = 0; s < 32; ++s) {
        a[s].x = hrow[s * 4];
        a[s].y = hrow[s * 4 + 1];
    }

    const int rOut = rowBase + (lane >> 4) * 8;          // C/D: VGPR r <-> M=r(+8)
    for (int ct = 0; ct < 8; ++ct) {
        const int col = ct * 16 + colLane;
        v8f acc = {};
        #pragma unroll
        for (int s = 0; s < 32; ++s) {
            v2f b;
            b.x = Wl[(s * 4 + kOff) * D1 + col];         // B frag: K rows striped like A
            b.y = Wl[(s * 4 + kOff + 1) * D1 + col];
            acc = __builtin_amdgcn_wmma_f32_16x16x4_f32(
                false, a[s], false, b, (short)0, acc, false, false);
        }
        float* outp = XP + (size_t)rOut * D1 + ct * 16 + colLane;
        #pragma unroll
        for (int r = 0; r < 8; ++r)
            outp[r * D1] = acc[r];
    }
}

// ---------- per-node attention logits: al_s/al_d [n,4] ----------
__global__ __launch_bounds__(256) void al_kernel(const float* __restrict__ XP,
                                                 const float* __restrict__ As,
                                                 const float* __restrict__ Ad,
                                                 float* __restrict__ als,
                                                 float* __restrict__ ald, int n) {
    int i = blockIdx.x * 256 + threadIdx.x;
    if (i >= n * HEADS) return;
    int node = i >> 2, h = i & 3;
    const float4* xv = (const float4*)(XP + (size_t)node * D1 + h * HID);
    const float4* sv = (const float4*)(As + h * HID);
    const float4* dv = (const float4*)(Ad + h * HID);
    float s = 0.f, d = 0.f;
    #pragma unroll
    for (int c = 0; c < HID / 4; ++c) {
        float4 x = xv[c], a = sv[c], b = dv[c];
        s += x.x * a.x + x.y * a.y + x.z * a.z + x.w * a.w;
        d += x.x * b.x + x.y * b.y + x.z * b.z + x.w * b.w;
    }
    als[i] = s;
    ald[i] = d;
}

// ---------- zero/neg-inf init of per-layer accumulators ----------
__global__ __launch_bounds__(256) void init_kernel(unsigned* __restrict__ emax,
                                                   float* __restrict__ den,
                                                   float* __restrict__ aggr, int n) {
    int i = blockIdx.x * 256 + threadIdx.x;
    if (i < n * HEADS) { emax[i] = 0x007FFFFFu; den[i] = 0.f; }  // enc(-inf)
    if (i < n * D1)    aggr[i] = 0.f;
}

// ---------- edge pass 1: segment max ----------
__global__ __launch_bounds__(256) void edge_max_kernel(const int* __restrict__ ei, int E, int EP,
                                                       const float* __restrict__ als,
                                                       const float* __restrict__ ald,
                                                       unsigned* __restrict__ emax) {
    int i = blockIdx.x * 256 + threadIdx.x;
    if (i >= EP * HEADS) return;
    int e = i >> 2, h = i & 3, s, d;
    get_edge(ei, E, e, s, d);
    float v = als[s * 4 + h] + ald[d * 4 + h];
    v = v > 0.f ? v : 0.2f * v;                          // leaky_relu(0.2)
    atomicMax(&emax[d * 4 + h], fenc(v));
}

// ---------- edge pass 2: segment sum of exp ----------
__global__ __launch_bounds__(256) void edge_expsum_kernel(const int* __restrict__ ei, int E, int EP,
                                                          const float* __restrict__ als,
                                                          const float* __restrict__ ald,
                                                          const unsigned* __restrict__ emax,
                                                          float* __restrict__ den) {
    int i = blockIdx.x * 256 + threadIdx.x;
    if (i >= EP * HEADS) return;
    int e = i >> 2, h = i & 3, s, d;
    get_edge(ei, E, e, s, d);
    float v = als[s * 4 + h] + ald[d * 4 + h];
    v = v > 0.f ? v : 0.2f * v;
    float m = fdec(emax[d * 4 + h]);
    atomicAdd(&den[d * 4 + h], expf(v - m));
}

// ---------- edge pass 3: alpha-weighted scatter (1 wave32 per edge) ----------
__global__ __launch_bounds__(256) void edge_scatter_kernel(const int* __restrict__ ei, int E, int EP,
                                                           const float* __restrict__ als,
                                                           const float* __restrict__ ald,
                                                           const unsigned* __restrict__ emax,
                                                           const float* __restrict__ den,
                                                           const float* __restrict__ XP,
                                                           float* __restrict__ aggr) {
    int gid = blockIdx.x * 256 + threadIdx.x;
    int e = gid >> 5;
    if (e >= EP) return;
    int lane = gid & 31;
    int head = lane >> 3;                                // 8 lanes per head
    int ch   = head * HID + (lane & 7) * 4;              // 4 channels per lane
    int s, d;
    get_edge(ei, E, e, s, d);
    float v = als[s * 4 + head] + ald[d * 4 + head];
    v = v > 0.f ? v : 0.2f * v;
    float m  = fdec(emax[d * 4 + head]);
    float ex = expf(v - m);
    float alpha = ex / (den[d * 4 + head] + 1e-16f);
    float4 xv = *(const float4*)(XP + (size_t)s * D1 + ch);
    float* op = aggr + (size_t)d * D1 + ch;
    atomicAdd(op + 0, xv.x * alpha);                     // resolves in the 192 MB L2
    atomicAdd(op + 1, xv.y * alpha);
    atomicAdd(op + 2, xv.z * alpha);
    atomicAdd(op + 3, xv.w * alpha);
}

// ---------- bias + ELU ----------
__global__ __launch_bounds__(256) void finalize_kernel(const float* __restrict__ aggr,
                                                       const float* __restrict__ b,
                                                       float* __restrict__ Hout, int n) {
    int i = blockIdx.x * 256 + threadIdx.x;
    if (i >= n * D1) return;
    float v = aggr[i] + b[i & 127];
    Hout[i] = v > 0.f ? v : expm1f(v);                   // elu, alpha=1
}

// ---------- classifier: h @ cW + cb -> sigmoid ----------
__global__ __launch_bounds__(256) void classifier_kernel(const float* __restrict__ H,
                                                         const float* __restrict__ cW,
                                                         const float* __restrict__ cb,
                                                         float* __restrict__ out, int n) {
    int node = blockIdx.x * 256 + threadIdx.x;
    if (node >= n) return;
    const float4* hv = (const float4*)(H + (size_t)node * D1);
    const float4* wv = (const float4*)cW;
    float acc = 0.f;
    #pragma unroll
    for (int c = 0; c < D1 / 4; ++c) {
        float4 h = hv[c], w = wv[c];
        acc += h.x * w.x + h.y * w.y + h.z * w.z + h.w * w.w;
    }
    acc += cb[0];
    out[node] = 1.f / (1.f + expf(-acc));
}

extern "C" void kernel_launch(void* const* d_in, const int* in_sizes, int n_in,
                              void* d_out, int out_size, void* d_ws, size_t ws_size,
                              hipStream_t stream) {
    const float* x  = (const float*)d_in[0];
    const int*   ei = (const int*)d_in[1];               // edge_index [2,E]
    const float* W[3]  = {(const float*)d_in[2], (const float*)d_in[6],  (const float*)d_in[10]};
    const float* AS[3] = {(const float*)d_in[3], (const float*)d_in[7],  (const float*)d_in[11]};
    const float* AD[3] = {(const float*)d_in[4], (const float*)d_in[8],  (const float*)d_in[12]};
    const float* B[3]  = {(const float*)d_in[5], (const float*)d_in[9],  (const float*)d_in[13]};
    const float* cW = (const float*)d_in[14];
    const float* cb = (const float*)d_in[15];

    const int n  = in_sizes[0] / 3;                      // 50000 (multiple of 16)
    const int E  = in_sizes[1] / 2;                      // 800000
    const int EP = E + n;                                // + self-loops

    // workspace carve-out (~80 MB, fits comfortably in L2 working set)
    float*    hb   = (float*)d_ws;                       // [n,128] activations
    float*    xp   = hb   + (size_t)n * D1;              // [n,128] pre-aggregation
    float*    aggr = xp   + (size_t)n * D1;              // [n,128] scatter accumulator
    float*    als  = aggr + (size_t)n * D1;              // [n,4]
    float*    ald  = als  + (size_t)n * HEADS;           // [n,4]
    unsigned* emax = (unsigned*)(ald + (size_t)n * HEADS);
    float*    den  = (float*)(emax + (size_t)n * HEADS);

    const dim3 blk(256);
    const int g_n128 = (n * D1 + 255) / 256;
    const int g_n4   = (n * HEADS + 255) / 256;
    const int g_e4   = (EP * HEADS + 255) / 256;
    const int g_ew   = (EP + 7) / 8;                     // 8 waves per block
    const int g_gemm = (n + 127) / 128;

    for (int L = 0; L < 3; ++L) {
        if (L == 0)
            gemm3_kernel<<<g_n128, blk, 0, stream>>>(x, W[0], xp, n);
        else
            gemm128_wmma_kernel<<<g_gemm, blk, 0, stream>>>(hb, W[L], xp, n);
        al_kernel<<<g_n4, blk, 0, stream>>>(xp, AS[L], AD[L], als, ald, n);
        init_kernel<<<g_n128, blk, 0, stream>>>(emax, den, aggr, n);
        edge_max_kernel<<<g_e4, blk, 0, stream>>>(ei, E, EP, als, ald, emax);
        edge_expsum_kernel<<<g_e4, blk, 0, stream>>>(ei, E, EP, als, ald, emax, den);
        edge_scatter_kernel<<<g_ew, blk, 0, stream>>>(ei, E, EP, als, ald, emax, den, xp, aggr);
        finalize_kernel<<<g_n128, blk, 0, stream>>>(aggr, B[L], hb, n);
    }
    classifier_kernel<<<(n + 255) / 256, blk, 0, stream>>>(hb, cW, cb, (float*)d_out, n);
}